// PreEncoder_88493506167404
// MI455X (gfx1250) — compile-verified
//
#include <hip/hip_runtime.h>
#include <hip/hip_bf16.h>

// ---------------------------------------------------------------------------
// PreEncoder (IEEE-754 fp16-style bit expansion): x[f32] -> 16 values in {-1,+1}
// per element. Streaming kernel: 16MiB in + 256MiB out  =>  ~12us @ 23.3 TB/s.
// gfx1250 paths used:
//   * TENSOR_LOAD_TO_LDS (TDM DMA) stages each block's 8KB input chunk in LDS,
//     waited with s_wait_tensorcnt.
//   * Non-temporal 128b global stores for the 256MB output stream (does not
//     fit the 192MB L2; written exactly once).
// ---------------------------------------------------------------------------

typedef __attribute__((ext_vector_type(4))) float        vf4;
typedef __attribute__((ext_vector_type(4))) unsigned int vu4;
typedef __attribute__((ext_vector_type(8))) int          vi8;
typedef __attribute__((ext_vector_type(4))) int          vi4;

#define BLOCK  256
#define ELEMS  8
#define CHUNK  (BLOCK * ELEMS)   // 2048 f32 = 8KB per block

__device__ __forceinline__ float bitval(unsigned int code, int k) {
    return ((code >> k) & 1u) ? 1.0f : -1.0f;
}

// Produce the 16 output values for one input float, exactly matching the
// reference's float math via integer bit manipulation:
//   bits = [s, e4..e0, m1..m10];  out = 2*bits - 1
__device__ __forceinline__ void encode16(float f, vf4& a, vf4& b, vf4& c, vf4& d) {
    unsigned int u    = __float_as_uint(f);
    unsigned int aexp = (u >> 23) & 0xFFu;
    unsigned int frac = u & 0x7FFFFFu;

    unsigned int code;
    float v0;
    if (aexp == 0xFFu) {
        // NaN: 0 11111 1000000000 ; +Inf: 0 11111 0 ; -Inf: 1 11111 0
        if (frac != 0u) code = 0x7E00u;
        else            code = (u & 0x80000000u) ? 0xFC00u : 0x7C00u;
        v0 = (code & 0x8000u) ? 1.0f : -1.0f;
    } else {
        int e_sci;
        unsigned int mant;
        if (aexp != 0u) {                    // normal f32
            e_sci = (int)aexp - 127;         // exact floor(log2|f|)
            mant  = frac >> 13;              // top 10 mantissa bits (truncate)
        } else if (frac == 0u) {             // +-0: ref forces e_sci = -15, m = 0
            e_sci = -15;
            mant  = 0u;
        } else {                             // denormal (never in this input, but exact)
            int p = 31 - __clz((int)frac);   // MSB position of frac
            e_sci = p - 149;
            mant  = (p >= 10) ? ((frac >> (p - 10)) & 0x3FFu)
                              : ((frac << (10 - p)) & 0x3FFu);
        }
        // e_bits[k] = floor((e_sci+15)/2^k) mod 2  ==  two's-complement bit k
        unsigned int ebits = ((unsigned int)(e_sci + 15)) & 31u;
        code = (ebits << 10) | mant;
        // sign via sign(f + 0.001): >0 -> bit 0 (-1), <0 -> bit 1 (+1), ==0 -> 0.0
        float sf = f + 0.001f;
        v0 = (sf > 0.0f) ? -1.0f : ((sf < 0.0f) ? 1.0f : 0.0f);
    }

    a.x = v0;
    a.y = bitval(code, 14); a.z = bitval(code, 13); a.w = bitval(code, 12);
    b.x = bitval(code, 11); b.y = bitval(code, 10); b.z = bitval(code,  9); b.w = bitval(code, 8);
    c.x = bitval(code,  7); c.y = bitval(code,  6); c.z = bitval(code,  5); c.w = bitval(code, 4);
    d.x = bitval(code,  3); d.y = bitval(code,  2); d.z = bitval(code,  1); d.w = bitval(code, 0);
}

__global__ __launch_bounds__(BLOCK) void
PreEncoder_88493506167404_kernel(const float* __restrict__ x,
                                 float* __restrict__ out, int n) {
    __shared__ float tile[CHUNK];
    const int block_start = (int)blockIdx.x * CHUNK;

    // ---- TDM: DMA this block's input chunk (global -> LDS), wave 0 only ----
    if (threadIdx.x < 32u) {
        unsigned long long gaddr =
            (unsigned long long)(uintptr_t)(x + block_start);
        unsigned int lds_off = (unsigned int)(uintptr_t)(&tile[0]); // addr[31:0] = LDS offset
        int remaining = n - block_start;
        unsigned int dim0 = (remaining < CHUNK) ? (unsigned int)remaining
                                                : (unsigned int)CHUNK;

        // D# group 0 (128b): count=1, lds_addr, global_addr[56:0], type=2
        vu4 g0;
        g0[0] = 1u;
        g0[1] = lds_off;
        g0[2] = (unsigned int)(gaddr & 0xFFFFFFFFull);
        g0[3] = (unsigned int)((gaddr >> 32) & 0x01FFFFFFull) | (2u << 30);

        // D# group 1 (256b): data_size=4B, 1D tensor dim0, 1D tile tile_dim0
        vi8 g1;
        g1[0] = (int)(2u << 16);                               // data_size = 2 (4 bytes)
        g1[1] = (int)((dim0 & 0xFFFFu) << 16);                 // tensor_dim0[15:0]
        g1[2] = (int)(((dim0 >> 16) & 0xFFFFu) | (1u << 16));  // tensor_dim0[31:16], tensor_dim1=1
        g1[3] = (int)(((unsigned)CHUNK & 0xFFFFu) << 16);      // tile_dim0 = CHUNK
        g1[4] = 0;                                             // tile_dim1/2 unused
        g1[5] = (int)CHUNK;                                    // tensor_dim0_stride
        g1[6] = 0;
        g1[7] = 0;

        vi4 g2 = {0, 0, 0, 0};           // D# group 2 (unused: <=2D tensor)
        vi4 g3 = {0, 0, 0, 0};           // D# group 3 (unused)
        vi8 g4 = {0, 0, 0, 0, 0, 0, 0, 0}; // extra group (clang-23 6-arg form)
        __builtin_amdgcn_tensor_load_to_lds(g0, g1, g2, g3, g4, 0);
        __builtin_amdgcn_s_wait_tensorcnt(0);
    }
    __syncthreads();

    // ---- compute + non-temporal streaming stores (lane-adjacent layout) ----
#pragma unroll
    for (int e = 0; e < ELEMS; ++e) {
        int local = e * BLOCK + (int)threadIdx.x;
        int gidx  = block_start + local;
        if (gidx < n) {
            float f = tile[local];
            vf4 a, b, c, d;
            encode16(f, a, b, c, d);
            float* dst = out + (size_t)gidx * 16;
            __builtin_nontemporal_store(a, (vf4*)(dst +  0));
            __builtin_nontemporal_store(b, (vf4*)(dst +  4));
            __builtin_nontemporal_store(c, (vf4*)(dst +  8));
            __builtin_nontemporal_store(d, (vf4*)(dst + 12));
        }
    }
}

extern "C" void kernel_launch(void* const* d_in, const int* in_sizes, int n_in,
                              void* d_out, int out_size, void* d_ws, size_t ws_size,
                              hipStream_t stream) {
    (void)n_in; (void)out_size; (void)d_ws; (void)ws_size;
    const float* x = (const float*)d_in[0];
    float* out = (float*)d_out;
    int n = in_sizes[0];                         // 64*4096*16 = 4,194,304
    int grid = (n + CHUNK - 1) / CHUNK;          // 2048 blocks
    PreEncoder_88493506167404_kernel<<<grid, BLOCK, 0, stream>>>(x, out, n);
}